// UnPooling2D_26749056319643
// MI455X (gfx1250) — compile-verified
//
#include <hip/hip_runtime.h>

// UnPooling2D (max-unpool, factor 2) for MI455X / gfx1250.
// Pure streaming scatter: bandwidth-bound (224 MiB traffic, ~10.1 us @ 23.3 TB/s).
// No matmul content -> WMMA intentionally not used; optimization levers are
// 128-bit coalesced VMEM, CDNA5 NT temporal hints, wave32-friendly sizing.

typedef float     v4f __attribute__((ext_vector_type(4)));
typedef long long v2l __attribute__((ext_vector_type(2)));

constexpr int B  = 16, H = 64, W = 64, C = 128;
constexpr int H2 = 2 * H, W2 = 2 * W;          // 128, 128
constexpr int VEC = 4;                          // channels per thread (one b128)
constexpr int CV  = C / VEC;                    // 32 vec4 chunks across C
constexpr int NTHREADS = B * H * W * CV;        // 2,097,152
constexpr int BLOCK = 256;                      // 8 wave32s

__global__ __launch_bounds__(BLOCK) void unpool2d_scatter(
    const float* __restrict__ pooled,
    const long long* __restrict__ indices,
    float* __restrict__ out)
{
    const int tid = blockIdx.x * BLOCK + threadIdx.x;

    // Decompose: tid = (((b*H + h)*W + w) * CV + cv)
    const int cv      = tid & (CV - 1);   // vec4 chunk in channel dim
    const int spatial = tid >> 5;         // (b*H + h)*W + w   (CV == 32)
    const int w  = spatial & (W - 1);
    const int bh = spatial >> 6;          // W == 64
    const int h  = bh & (H - 1);
    const int b  = bh >> 6;               // H == 64
    const int c  = cv * VEC;

    const int inElem = ((b * H + h) * W + w) * C + c;

    // Touch-once streams: non-temporal loads (don't pollute 192MB L2).
    const v4f v   = __builtin_nontemporal_load((const v4f*)(pooled + inElem));
    const v2l i01 = __builtin_nontemporal_load((const v2l*)(indices + inElem));
    const v2l i23 = __builtin_nontemporal_load((const v2l*)(indices + inElem + 2));

    // idx = ((2h+dy)*W2 + (2w+dx))*C + c  with C = W2 = 128:
    //   dx = (idx >> 7) & 1  (2w is even),  dy = (idx >> 14) & 1
    // Quadrant code q = 2*dy + dx in [0,4).
    const int q0 = (((int)(i01.x >> 14) & 1) << 1) | ((int)(i01.x >> 7) & 1);
    const int q1 = (((int)(i01.y >> 14) & 1) << 1) | ((int)(i01.y >> 7) & 1);
    const int q2 = (((int)(i23.x >> 14) & 1) << 1) | ((int)(i23.x >> 7) & 1);
    const int q3 = (((int)(i23.y >> 14) & 1) << 1) | ((int)(i23.y >> 7) & 1);

    v4f o[4];
#pragma unroll
    for (int k = 0; k < 4; ++k) {
        o[k].x = (q0 == k) ? v.x : 0.0f;
        o[k].y = (q1 == k) ? v.y : 0.0f;
        o[k].z = (q2 == k) ? v.z : 0.0f;
        o[k].w = (q3 == k) ? v.w : 0.0f;
    }

    // Four 2x2-window positions; each store is a contiguous 512B burst across
    // the 32 lanes of the wave (lanes advance along C). Covers out[] fully, so
    // no pre-zeroing of d_out is needed.
    const int outBase = ((b * H2 + 2 * h) * W2 + 2 * w) * C + c;
    __builtin_nontemporal_store(o[0], (v4f*)(out + outBase));                 // (2h  , 2w  )
    __builtin_nontemporal_store(o[1], (v4f*)(out + outBase + C));             // (2h  , 2w+1)
    __builtin_nontemporal_store(o[2], (v4f*)(out + outBase + W2 * C));        // (2h+1, 2w  )
    __builtin_nontemporal_store(o[3], (v4f*)(out + outBase + W2 * C + C));    // (2h+1, 2w+1)
}

extern "C" void kernel_launch(void* const* d_in, const int* in_sizes, int n_in,
                              void* d_out, int out_size, void* d_ws, size_t ws_size,
                              hipStream_t stream) {
    (void)in_sizes; (void)n_in; (void)out_size; (void)d_ws; (void)ws_size;
    const float*     pooled  = (const float*)d_in[0];
    const long long* indices = (const long long*)d_in[1];  // reference casts to int64
    float*           out     = (float*)d_out;

    static_assert(NTHREADS % BLOCK == 0, "grid sizing");
    dim3 grid(NTHREADS / BLOCK);   // 8192 blocks
    dim3 block(BLOCK);
    unpool2d_scatter<<<grid, block, 0, stream>>>(pooled, indices, out);
}